// RPNNew_13597866459571
// MI455X (gfx1250) — compile-verified
//
#include <hip/hip_runtime.h>
#include <stdint.h>

#define BATCH 8
#define NTOT  242991
#define KTOT  8741
#define NEGF  (-1e30f)
#define XCLIP 4.135166556742356f   /* log(1000/16) */

typedef int v4i_vec __attribute__((__vector_size__(4 * sizeof(int))));

// ---------------------------------------------------------------------------
// CDNA5 async global->LDS helpers (gfx1250). Probe-confirmed signatures:
//   b32 : (int  addrspace(1)*, int  addrspace(3)*, imm, imm)
//   b128: (v4i  addrspace(1)*, v4i  addrspace(3)*, imm, imm)
// ---------------------------------------------------------------------------
__device__ __forceinline__ void async_g2l_b32(void* lds, const void* g) {
#if __has_builtin(__builtin_amdgcn_global_load_async_to_lds_b32)
  int* gp = (int*)g;   // C-style cast drops const
  int* lp = (int*)lds;
  __builtin_amdgcn_global_load_async_to_lds_b32(
      (__attribute__((address_space(1))) int*)gp,
      (__attribute__((address_space(3))) int*)lp,
      0, 0);
#else
  *(int*)lds = *(const int*)g;   // synchronous fallback
#endif
}

__device__ __forceinline__ void async_g2l_b128(void* lds, const void* g) {
#if __has_builtin(__builtin_amdgcn_global_load_async_to_lds_b128)
  v4i_vec* gp = (v4i_vec*)g;
  v4i_vec* lp = (v4i_vec*)lds;
  __builtin_amdgcn_global_load_async_to_lds_b128(
      (__attribute__((address_space(1))) v4i_vec*)gp,
      (__attribute__((address_space(3))) v4i_vec*)lp,
      0, 0);
#else
  async_g2l_b32((int*)lds + 0, (const int*)g + 0);
  async_g2l_b32((int*)lds + 1, (const int*)g + 1);
  async_g2l_b32((int*)lds + 2, (const int*)g + 2);
  async_g2l_b32((int*)lds + 3, (const int*)g + 3);
#endif
}

__device__ __forceinline__ void wait_async0() {
#if __has_builtin(__builtin_amdgcn_s_wait_asynccnt)
  __builtin_amdgcn_s_wait_asynccnt(0);
#else
  asm volatile("s_wait_asynccnt 0" ::: "memory");
#endif
}

// monotonic float -> uint key (order preserving)
__device__ __forceinline__ unsigned fkey(float f) {
  unsigned u = __float_as_uint(f);
  return (u & 0x80000000u) ? ~u : (u | 0x80000000u);
}

// ---------------------------------------------------------------------------
// Kernel 1: fused permute_and_flatten + box decode + clip + min-size validity
// ---------------------------------------------------------------------------
__global__ __launch_bounds__(256) void decode_kernel(
    const float* __restrict__ c0, const float* __restrict__ c1,
    const float* __restrict__ c2, const float* __restrict__ c3,
    const float* __restrict__ c4,
    const float* __restrict__ b0, const float* __restrict__ b1,
    const float* __restrict__ b2, const float* __restrict__ b3,
    const float* __restrict__ b4,
    const float* __restrict__ anchors,
    float* __restrict__ sraw, float* __restrict__ snms,
    float* __restrict__ boxes) {
  int t = blockIdx.x * blockDim.x + threadIdx.x;
  if (t >= BATCH * NTOT) return;
  int b = t / NTOT;
  int i = t - b * NTOT;

  const float* cls; const float* box; int rem, H, W;
  if      (i < 182400) { cls = c0; box = b0; rem = i;          H = 200; W = 304; }
  else if (i < 228000) { cls = c1; box = b1; rem = i - 182400; H = 100; W = 152; }
  else if (i < 239400) { cls = c2; box = b2; rem = i - 228000; H =  50; W =  76; }
  else if (i < 242250) { cls = c3; box = b3; rem = i - 239400; H =  25; W =  38; }
  else                 { cls = c4; box = b4; rem = i - 242250; H =  13; W =  19; }

  int a  = rem % 3;
  int hw = rem / 3;
  int w  = hw % W;
  int h  = hw / W;
  int plane = H * W;
  int pix   = h * W + w;

  float score = cls[(size_t)(b * 3 + a) * plane + pix];
  const float* bp = box + (size_t)(b * 12 + a * 4) * plane + pix;
  float dx = bp[0];
  float dy = bp[(size_t)plane];
  float dw = bp[(size_t)2 * plane];
  float dh = bp[(size_t)3 * plane];

  const float* an = anchors + (size_t)i * 4;
  float ax1 = an[0], ay1 = an[1], ax2 = an[2], ay2 = an[3];
  float wa = ax2 - ax1, ha = ay2 - ay1;
  float cxa = ax1 + 0.5f * wa, cya = ay1 + 0.5f * ha;
  dw = fminf(dw, XCLIP);
  dh = fminf(dh, XCLIP);
  float cx = dx * wa + cxa, cy = dy * ha + cya;
  float pw = expf(dw) * wa, ph = expf(dh) * ha;
  float x1 = cx - 0.5f * pw, y1 = cy - 0.5f * ph;
  float x2 = cx + 0.5f * pw, y2 = cy + 0.5f * ph;
  x1 = fminf(fmaxf(x1, 0.f), 1216.f);
  x2 = fminf(fmaxf(x2, 0.f), 1216.f);
  y1 = fminf(fmaxf(y1, 0.f),  800.f);
  y2 = fminf(fmaxf(y2, 0.f),  800.f);
  bool valid = ((x2 - x1) >= 1e-3f) && ((y2 - y1) >= 1e-3f);

  size_t o = (size_t)b * NTOT + i;
  sraw[o] = score;
  snms[o] = valid ? score : NEGF;
  float* op = boxes + o * 4;
  op[0] = x1; op[1] = y1; op[2] = x2; op[3] = y2;
}

// ---------------------------------------------------------------------------
// Kernel 2: exact top-k per (batch, level), 4-pass MSB-first radix select.
// ---------------------------------------------------------------------------
#define TIECAP 4096

__global__ __launch_bounds__(1024) void topk_kernel(
    const float* __restrict__ sraw, int* __restrict__ cand) {
  const int LOFF[5] = {0, 182400, 228000, 239400, 242250};
  const int LSZ[5]  = {182400, 45600, 11400, 2850, 741};
  const int KL[5]   = {2000, 2000, 2000, 2000, 741};
  const int CB[5]   = {0, 2000, 4000, 6000, 8000};

  int blk = blockIdx.x;
  int b = blk / 5, lev = blk % 5;
  int n = LSZ[lev], k = KL[lev];
  const float* s = sraw + (size_t)b * NTOT + LOFF[lev];
  int* out = cand + b * KTOT + CB[lev];

  __shared__ unsigned hist[256];
  __shared__ unsigned s_prefix;
  __shared__ int s_need;
  __shared__ unsigned c_gt, c_eq;
  __shared__ int tiebuf[TIECAP];

  if (threadIdx.x == 0) { s_prefix = 0u; s_need = k; c_gt = 0u; c_eq = 0u; }

  for (int p = 0; p < 4; ++p) {
    __syncthreads();
    for (int x = threadIdx.x; x < 256; x += blockDim.x) hist[x] = 0u;
    __syncthreads();
    unsigned prefix = s_prefix;
    int sh = 24 - 8 * p;
    for (int i = threadIdx.x; i < n; i += blockDim.x) {
      unsigned u = fkey(s[i]);
      bool m = (p == 0) || ((u >> (sh + 8)) == prefix);
      if (m) atomicAdd(&hist[(u >> sh) & 0xFFu], 1u);
    }
    __syncthreads();
    if (threadIdx.x == 0) {
      int need = s_need;
      unsigned pref = s_prefix;
      for (int d = 255; d >= 0; --d) {
        int c = (int)hist[d];
        if (need <= c) { s_prefix = (pref << 8) | (unsigned)d; break; }
        need -= c;
      }
      s_need = need;
    }
  }
  __syncthreads();
  unsigned uthr = s_prefix;
  int need = s_need;            // #elements equal to threshold to keep
  int gt_total = k - need;      // #elements strictly greater

  for (int i = threadIdx.x; i < n; i += blockDim.x) {
    unsigned u = fkey(s[i]);
    if (u > uthr) {
      unsigned pidx = atomicAdd(&c_gt, 1u);
      out[pidx] = i;
    } else if (u == uthr) {
      unsigned pidx = atomicAdd(&c_eq, 1u);
      if (pidx < TIECAP) tiebuf[pidx] = i;
    }
  }
  __syncthreads();
  int c = (int)c_eq; if (c > TIECAP) c = TIECAP;
  if ((int)c_eq <= TIECAP) {
    // deterministic: rank ties by ascending anchor index
    for (int x = threadIdx.x; x < c; x += blockDim.x) {
      int mine = tiebuf[x];
      int rank = 0;
      for (int y = 0; y < c; ++y) rank += (tiebuf[y] < mine) ? 1 : 0;
      if (rank < need) out[gt_total + rank] = mine;
    }
  } else {
    for (int x = threadIdx.x; x < need; x += blockDim.x) out[gt_total + x] = tiebuf[x];
  }
}

// ---------------------------------------------------------------------------
// Kernel 3: batched greedy NMS, one workgroup (1024 threads = 32 waves) per
// image. Entire 8741-candidate set resident in ~210 KB of the 320 KB WGP LDS.
// All staging (index table, gathered boxes, gathered scores) flows through
// CDNA5 async global->LDS DMA (ASYNCcnt / s_wait_asynccnt). Argmax via
// ds_max_u64 on a 64-bit key (score | ~anchor_idx | slot): deterministic,
// matches the reference's stable sort tie-breaking.
// ---------------------------------------------------------------------------
__global__ __launch_bounds__(1024) void nms_kernel(
    const float* __restrict__ boxes, const float* __restrict__ sraw,
    const float* __restrict__ snms, const int* __restrict__ cand,
    float* __restrict__ out) {
  __shared__ __align__(16) float sbox[KTOT][4];
  __shared__ float ss[KTOT];
  __shared__ int   sidx[KTOT];
  __shared__ unsigned long long best;

  int b   = blockIdx.x;
  int tid = threadIdx.x;
  const int* candb = cand + b * KTOT;

  // Phase 1: async DMA the candidate index table into LDS.
  for (int j = tid; j < KTOT; j += blockDim.x)
    async_g2l_b32(&sidx[j], &candb[j]);
  wait_async0();
  __syncthreads();

  // Phase 2: async gather boxes (b128) and nms scores (b32) into LDS.
  for (int j = tid; j < KTOT; j += blockDim.x) {
    int gi = sidx[j];
    async_g2l_b128(&sbox[j][0], boxes + ((size_t)b * NTOT + gi) * 4);
    async_g2l_b32(&ss[j], &snms[(size_t)b * NTOT + gi]);
  }
  wait_async0();
  __syncthreads();

  // Phase 3: apply per-level batched-NMS offsets in place.
  for (int j = tid; j < KTOT; j += blockDim.x) {
    int lev = (j < 2000) ? 0 : (j < 4000) ? 1 : (j < 6000) ? 2 : (j < 8000) ? 3 : 4;
    float off = (float)lev * 1217.0f;   // max(H,W)+1
    sbox[j][0] += off; sbox[j][1] += off;
    sbox[j][2] += off; sbox[j][3] += off;
  }
  __syncthreads();

  for (int t = 0; t < 100; ++t) {
    __syncthreads();                      // prior suppression/reads complete
    if (tid == 0) best = 0ull;
    __syncthreads();

    unsigned long long lk = 0ull;
    for (int j = tid; j < KTOT; j += blockDim.x) {
      // 32b score key | 18b inverted anchor idx (smaller idx wins) | 14b slot
      unsigned long long key =
          ((unsigned long long)fkey(ss[j]) << 32) |
          ((unsigned long long)(unsigned)(262143 - sidx[j]) << 14) |
          (unsigned long long)(unsigned)j;
      lk = (key > lk) ? key : lk;
    }
    atomicMax(&best, lk);                 // ds_max_u64
    __syncthreads();

    unsigned long long bk = best;
    int pi = (int)(bk & 0x3FFFull);
    bool ok = ss[pi] > (NEGF * 0.5f);

    if (ok) {
      float bx1 = sbox[pi][0], by1 = sbox[pi][1];
      float bx2 = sbox[pi][2], by2 = sbox[pi][3];
      float barea = (bx2 - bx1) * (by2 - by1);
      for (int j = tid; j < KTOT; j += blockDim.x) {
        float jx1 = sbox[j][0], jy1 = sbox[j][1];
        float jx2 = sbox[j][2], jy2 = sbox[j][3];
        float xx1 = fmaxf(jx1, bx1), yy1 = fmaxf(jy1, by1);
        float xx2 = fminf(jx2, bx2), yy2 = fminf(jy2, by2);
        float inter = fmaxf(xx2 - xx1, 0.f) * fmaxf(yy2 - yy1, 0.f);
        float aj = (jx2 - jx1) * (jy2 - jy1);
        float un = aj + barea - inter;
        float iou = (un > 0.f) ? (inter / un) : 0.f;
        if (iou > 0.7f || j == pi) ss[j] = NEGF;
      }
    }
    if (tid == 0) {
      float* ob = out + ((size_t)b * 100 + t) * 4;
      float* os = out + (size_t)BATCH * 100 * 4 + b * 100 + t;
      if (ok) {
        int gi = sidx[pi];
        const float* bp = boxes + ((size_t)b * NTOT + gi) * 4;
        ob[0] = bp[0]; ob[1] = bp[1]; ob[2] = bp[2]; ob[3] = bp[3];
        *os = sraw[(size_t)b * NTOT + gi];
      } else {
        ob[0] = 0.f; ob[1] = 0.f; ob[2] = 0.f; ob[3] = 0.f;
        *os = 0.f;
      }
    }
  }
}

// ---------------------------------------------------------------------------
// Host launcher
// ---------------------------------------------------------------------------
static inline size_t align256(size_t x) { return (x + 255) & ~(size_t)255; }

extern "C" void kernel_launch(void* const* d_in, const int* in_sizes, int n_in,
                              void* d_out, int out_size, void* d_ws, size_t ws_size,
                              hipStream_t stream) {
  (void)in_sizes; (void)n_in; (void)out_size; (void)ws_size;
  // setup_inputs dict order: cls0, box0, cls1, box1, ..., cls4, box4, anchors
  const float* cls[5]; const float* box[5];
  for (int i = 0; i < 5; ++i) {
    cls[i] = (const float*)d_in[2 * i];
    box[i] = (const float*)d_in[2 * i + 1];
  }
  const float* anchors = (const float*)d_in[10];

  char* ws = (char*)d_ws;
  size_t BN = (size_t)BATCH * NTOT;
  size_t o = 0;
  float* sraw  = (float*)(ws + o); o += align256(BN * 4);
  float* snms  = (float*)(ws + o); o += align256(BN * 4);
  float* boxes = (float*)(ws + o); o += align256(BN * 16);
  int*   cand  = (int*)  (ws + o); o += align256((size_t)BATCH * KTOT * 4);

  int total = BATCH * NTOT;
  decode_kernel<<<(total + 255) / 256, 256, 0, stream>>>(
      cls[0], cls[1], cls[2], cls[3], cls[4],
      box[0], box[1], box[2], box[3], box[4],
      anchors, sraw, snms, boxes);

  topk_kernel<<<BATCH * 5, 1024, 0, stream>>>(sraw, cand);

  nms_kernel<<<BATCH, 1024, 0, stream>>>(boxes, sraw, snms, cand, (float*)d_out);
}